// CUDAPagedMultiHeadAttention_6098853560978
// MI455X (gfx1250) — compile-verified
//
#include <hip/hip_runtime.h>
#include <math.h>

#define D_MODEL   4096
#define NUM_HEADS 32
#define HEAD_DIM  128
#define BATCH     8
#define BLK_SZ    16
#define MAX_BLKS  128
#define T_MAX     (MAX_BLKS * BLK_SZ)   // 2048

typedef float v2f __attribute__((ext_vector_type(2)));
typedef float v8f __attribute__((ext_vector_type(8)));

// ---------------------------------------------------------------------------
// 16-wide output tile of Y = X @ W^T  (X: [8 x 4096] padded to M=16, W row-major)
// Uses V_WMMA_F32_16X16X4_F32 (full fp32 precision, 8 VGPR f32 accumulator).
//
// A layout (32-bit A, 16x4): lanes 0-15: M=lane,   V0=K0, V1=K1
//                            lanes16-31: M=lane-16,V0=K2, V1=K3
// B layout (4x16, mirrored):  lanes 0-15: N=lane,   V0=K0, V1=K1
//                            lanes16-31: N=lane-16,V0=K2, V1=K3
// C/D layout: VGPR r -> lanes 0-15: (M=r, N=lane); lanes 16-31: (M=r+8) [padding]
//
// blockDim = 128 (4 waves); waves split K (1024 each), LDS-reduce the C tiles.
// Wave index is read-first-laned so the k-loop is scalar (EXEC stays all-1s
// around the WMMAs, no exec-mask loop).
// ---------------------------------------------------------------------------
__device__ __forceinline__ void gemm16_tile(const float* __restrict__ X,
                                            const float* __restrict__ W,
                                            float* __restrict__ Y,
                                            int n0)
{
    __shared__ float red[4][32][8];
    const int tid  = threadIdx.x;
    const int wave = __builtin_amdgcn_readfirstlane(tid >> 5);  // SGPR: scalar loop ctl
    const int lane = tid & 31;
    const int mn   = lane & 15;
    const int kofs = (lane >> 4) * 2;          // 0 or 2

    const int   mclamp = (mn < 8) ? mn : 7;    // rows 8..15 are padding
    const float* xrow  = X + (size_t)mclamp * D_MODEL;
    const float* wrow  = W + (size_t)(n0 + mn) * D_MODEL;

    v8f c = {};
    const int kbeg = wave * (D_MODEL / 4);     // 256 WMMA steps per wave
    const int kend = kbeg + (D_MODEL / 4);
    #pragma unroll 4
    for (int k = kbeg; k < kend; k += 4) {
        float2 xa = *(const float2*)(xrow + k + kofs);
        float2 wb = *(const float2*)(wrow + k + kofs);
        v2f a, b;
        a.x = (mn < 8) ? xa.x : 0.0f;          // zero padded batch rows
        a.y = (mn < 8) ? xa.y : 0.0f;
        b.x = wb.x;
        b.y = wb.y;
        // 8 args: (neg_a, A, neg_b, B, c_mod, C, reuse_a, reuse_b)
        c = __builtin_amdgcn_wmma_f32_16x16x4_f32(false, a, false, b,
                                                  (short)0, c, false, false);
    }

    #pragma unroll
    for (int r = 0; r < 8; ++r) red[wave][lane][r] = c[r];
    __syncthreads();

    // lanes 0-15 of wave 0 hold the 8 real batch rows (M=0..7)
    if (tid < 16) {
        #pragma unroll
        for (int r = 0; r < 8; ++r) {
            float s = red[0][tid][r] + red[1][tid][r] + red[2][tid][r] + red[3][tid][r];
            Y[(size_t)r * D_MODEL + n0 + tid] = s;
        }
    }
}

// blockIdx.x in [0, 768): mat = /256 selects Wq/Wk/Wv, tile = %256 selects 16 outputs
__global__ __launch_bounds__(128)
void qkv_kernel(const float* __restrict__ x,
                const float* __restrict__ Wq,
                const float* __restrict__ Wk,
                const float* __restrict__ Wv,
                float* __restrict__ qkv_ws)
{
    const int mat  = blockIdx.x >> 8;
    const int tile = blockIdx.x & 255;
    const float* W = (mat == 0) ? Wq : (mat == 1) ? Wk : Wv;
    float* Y = qkv_ws + (size_t)mat * (BATCH * D_MODEL);
    gemm16_tile(x, W, Y, tile * 16);
}

__global__ __launch_bounds__(128)
void oproj_kernel(const float* __restrict__ ctx,
                  const float* __restrict__ Wo,
                  float* __restrict__ out)
{
    gemm16_tile(ctx, Wo, out, blockIdx.x * 16);
}

// ---------------------------------------------------------------------------
// Paged attention for one (batch, head). Tokens [0, pos) stream from the
// cache branch-free; the current token (t == pos) is peeled out and served
// from the freshly projected k/v in workspace (functional cache update:
// block_tables entries are unique, so no older token aliases that slot).
// ---------------------------------------------------------------------------
__global__ __launch_bounds__(256)
void attn_kernel(const float* __restrict__ q_ws,
                 const float* __restrict__ k_ws,
                 const float* __restrict__ v_ws,
                 const float* __restrict__ key_cache,
                 const float* __restrict__ value_cache,
                 const int*   __restrict__ block_tables,
                 const int*   __restrict__ positions,
                 float* __restrict__ ctx_ws)
{
    __shared__ float scores[T_MAX];
    __shared__ float qs[HEAD_DIM];
    __shared__ float red[256];
    __shared__ float ctxp[2][HEAD_DIM];

    const int tid = threadIdx.x;
    const int b   = blockIdx.x >> 5;
    const int h   = blockIdx.x & 31;
    const int pos = positions[b];
    const int T   = pos + 1;
    const float scale = 0.08838834764831845f;  // 1/sqrt(128)

    if (tid < HEAD_DIM) qs[tid] = q_ws[(size_t)b * D_MODEL + h * HEAD_DIM + tid];
    __syncthreads();

    // ---- Phase A: scores over cached tokens [0, pos), branch-free ----
    float lmax = -INFINITY;
    for (int t = tid; t < pos; t += 256) {
        int blk = block_tables[b * MAX_BLKS + (t >> 4)];
        const float* kp = key_cache +
            ((size_t)(blk * BLK_SZ + (t & 15)) * NUM_HEADS + h) * HEAD_DIM;
        float s = 0.0f;
        const float4* kp4 = (const float4*)kp;
        #pragma unroll 4
        for (int i = 0; i < HEAD_DIM / 4; ++i) {
            float4 kv = kp4[i];
            s += kv.x * qs[4*i+0] + kv.y * qs[4*i+1] + kv.z * qs[4*i+2] + kv.w * qs[4*i+3];
        }
        s *= scale;
        scores[t] = s;
        lmax = fmaxf(lmax, s);
    }
    // peeled current token from workspace (thread 0)
    if (tid == 0) {
        const float4* kp4 = (const float4*)(k_ws + (size_t)b * D_MODEL + h * HEAD_DIM);
        float s = 0.0f;
        #pragma unroll 4
        for (int i = 0; i < HEAD_DIM / 4; ++i) {
            float4 kv = kp4[i];
            s += kv.x * qs[4*i+0] + kv.y * qs[4*i+1] + kv.z * qs[4*i+2] + kv.w * qs[4*i+3];
        }
        s *= scale;
        scores[pos] = s;
        lmax = fmaxf(lmax, s);
    }

    // ---- Phase B: block max, exp, sum ----
    red[tid] = lmax;
    __syncthreads();
    for (int s = 128; s > 0; s >>= 1) {
        if (tid < s) red[tid] = fmaxf(red[tid], red[tid + s]);
        __syncthreads();
    }
    const float gmax = red[0];
    __syncthreads();

    float lsum = 0.0f;
    for (int t = tid; t < T; t += 256) {
        float e = __expf(scores[t] - gmax);
        scores[t] = e;
        lsum += e;
    }
    red[tid] = lsum;
    __syncthreads();
    for (int s = 128; s > 0; s >>= 1) {
        if (tid < s) red[tid] += red[tid + s];
        __syncthreads();
    }
    const float inv = 1.0f / red[0];

    // ---- Phase C: ctx accumulation (2 groups x 128 lanes, coalesced V reads),
    //      cached tokens [0, pos) branch-free, current token peeled ----
    const int grp = tid >> 7;
    const int d   = tid & 127;
    float acc = 0.0f;
    for (int t = grp; t < pos; t += 2) {
        int blk = block_tables[b * MAX_BLKS + (t >> 4)];
        float vv = value_cache[((size_t)(blk * BLK_SZ + (t & 15)) * NUM_HEADS + h) * HEAD_DIM + d];
        acc += scores[t] * vv;
    }
    if (grp == (pos & 1))   // token pos belongs to group (pos % 2)
        acc += scores[pos] * v_ws[(size_t)b * D_MODEL + h * HEAD_DIM + d];
    ctxp[grp][d] = acc;
    __syncthreads();

    if (tid < HEAD_DIM)
        ctx_ws[(size_t)b * D_MODEL + h * HEAD_DIM + tid] =
            (ctxp[0][tid] + ctxp[1][tid]) * inv;
}

// ---------------------------------------------------------------------------
extern "C" void kernel_launch(void* const* d_in, const int* in_sizes, int n_in,
                              void* d_out, int out_size, void* d_ws, size_t ws_size,
                              hipStream_t stream)
{
    const float* x   = (const float*)d_in[0];
    const float* Wq  = (const float*)d_in[1];
    const float* Wk  = (const float*)d_in[2];
    const float* Wv  = (const float*)d_in[3];
    const float* Wo  = (const float*)d_in[4];
    const float* kc  = (const float*)d_in[5];
    const float* vc  = (const float*)d_in[6];
    const int*   bt  = (const int*)d_in[7];
    const int*   pos = (const int*)d_in[8];

    float* out = (float*)d_out;
    float* ws  = (float*)d_ws;

    float* q_ws   = ws;                 //  8 x 4096
    float* k_ws   = ws + 32768;         //  8 x 4096
    float* v_ws   = ws + 65536;         //  8 x 4096
    float* ctx_ws = ws + 98304;         //  8 x 4096   (total 512 KB)

    // 3 matrices x 256 tiles, 4 K-split waves each
    qkv_kernel<<<dim3(768), dim3(128), 0, stream>>>(x, Wq, Wk, Wv, q_ws);
    // one block per (batch, head)
    attn_kernel<<<dim3(256), dim3(256), 0, stream>>>(q_ws, k_ws, v_ws, kc, vc, bt, pos, ctx_ws);
    // 256 output tiles
    oproj_kernel<<<dim3(256), dim3(128), 0, stream>>>(ctx_ws, Wo, out);
}